// sign_pose_74680891343463
// MI455X (gfx1250) — compile-verified
//
#include <hip/hip_runtime.h>
#include <hip/hip_bf16.h>
#include <stdint.h>

#define H_IMG 1080
#define W_IMG 1920
#define NJ    26
#define NPAF  52
#define NCH   25          // NJOINT-1 blurred channels
#define KTOP  32
#define RAD   12          // truncate*sigma rounded
#define SEG_PIX 2048
#define NSEG  1013        // ceil(1080*1920 / 2048)
#define NCAND (NSEG * KTOP)   // 32416 candidates per channel
#define VROWS 88          // max vertical window rows per workgroup (64 + 24)

typedef __attribute__((ext_vector_type(2))) float    v2f;
typedef __attribute__((ext_vector_type(8))) float    v8f;
typedef __attribute__((ext_vector_type(4))) unsigned v4u;
typedef __attribute__((ext_vector_type(8))) int      v8i;
typedef __attribute__((ext_vector_type(4))) int      v4i;

// Zero-padded 25-tap Gaussian (sigma=3). Band index d in [-15, 40] maps to
// GWPAD[d + 16]; out-of-band entries are 0 so no branch is needed.
__constant__ float GWPAD[72] = {
    0.f, 0.f, 0.f, 0.f, 0.f, 0.f, 0.f, 0.f, 0.f, 0.f, 0.f, 0.f, 0.f, 0.f, 0.f, 0.f,
    4.46110e-05f, 1.60095e-04f, 5.14108e-04f, 1.47730e-03f, 3.79880e-03f,
    8.74090e-03f, 1.79976e-02f, 3.31600e-02f, 5.46716e-02f, 8.06592e-02f,
    1.06485690e-01f, 1.25797950e-01f, 1.32984520e-01f, 1.25797950e-01f,
    1.06485690e-01f, 8.06592e-02f, 5.46716e-02f, 3.31600e-02f, 1.79976e-02f,
    8.74090e-03f, 3.79880e-03f, 1.47730e-03f, 5.14108e-04f, 1.60095e-04f,
    4.46110e-05f,
    0.f, 0.f, 0.f, 0.f, 0.f, 0.f, 0.f, 0.f, 0.f, 0.f, 0.f, 0.f, 0.f, 0.f, 0.f,
    0.f, 0.f, 0.f, 0.f, 0.f, 0.f, 0.f, 0.f, 0.f, 0.f, 0.f, 0.f, 0.f, 0.f, 0.f, 0.f};

__constant__ int LIMB_A[24] = {1,1,2,3,1,5,6,1,8,9,10, 8,12,13, 0, 0,15,16,11,11,14,14,22,19};
__constant__ int LIMB_B[24] = {0,2,3,4,5,6,7,8,9,10,11,12,13,14,15,16,17,18,24,22,21,19,23,20};
__constant__ int MAPX[24]   = {30,14,16,18,22,24,26,0,6,2,4, 8,10,12,32,34,36,38,50,46,44,40,48,42};
__constant__ int MAPY[24]   = {31,15,17,19,23,25,27,1,7,3,5, 9,11,13,33,35,37,39,51,47,45,41,49,43};

__device__ __forceinline__ int reflect_idx(int i, int n) {
    i = (i < 0) ? (-i - 1) : i;            // symmetric padding: -1 -> 0, -2 -> 1
    return (i >= n) ? (2 * n - 1 - i) : i; // n -> n-1, n+1 -> n-2
}

__device__ __forceinline__ float bandw(int d) { return GWPAD[d + 16]; }

// TDM builtin availability / arity selection (5-arg ROCm7.2 vs 6-arg therock).
#if defined(__has_builtin)
#if __has_builtin(__builtin_amdgcn_tensor_load_to_lds)
#define HAVE_TDM 1
#endif
#endif
#if defined(HAVE_TDM)
#if __has_include(<hip/amd_detail/amd_gfx1250_TDM.h>)
#define TDM_LOAD(g0, g1, g2, g3)                                               \
    __builtin_amdgcn_tensor_load_to_lds((g0), (g1), (g2), (g3),                \
                                        (v8i){0, 0, 0, 0, 0, 0, 0, 0}, 0)
#else
#define TDM_LOAD(g0, g1, g2, g3)                                               \
    __builtin_amdgcn_tensor_load_to_lds((g0), (g1), (g2), (g3), 0)
#endif
#endif

// ---------------------------------------------------------------------------
// Horizontal blur: one wave per 16x16 output tile of one channel.
// C(16x16) = A(16x40 input pixels) x B(40x16 banded Gaussian), 10 x WMMA K=4.
// Interior tiles use one per-lane base pointer + constant offsets; only the
// two border tile columns pay for symmetric reflection.
// ---------------------------------------------------------------------------
__global__ void blur_h_kernel(const float* __restrict__ hm, float* __restrict__ tmp) {
    const int lane = threadIdx.x;          // 0..31 (wave32)
    const int m    = lane & 15;            // M row (A) / N col (B,C)
    const int half = lane >> 4;
    const int x0 = blockIdx.x * 16;
    const int y0 = blockIdx.y * 16;
    const int c  = blockIdx.z;

    const int    ry      = reflect_idx(y0 + m, H_IMG);
    const size_t rowbase = (size_t)ry * W_IMG;

    v8f acc = {};
    if (x0 >= RAD && x0 + 28 <= W_IMG) {
        // Interior: linear addressing, immediate offsets s*4*NJ(+NJ).
        const float* base = hm + (rowbase + (size_t)(x0 - RAD + half * 2)) * NJ + c;
#pragma unroll
        for (int s = 0; s < 10; ++s) {
            const int kb = 4 * s + half * 2;
            v2f a;
            a.x = base[s * 4 * NJ];
            a.y = base[s * 4 * NJ + NJ];
            v2f b;
            b.x = bandw(kb - m);
            b.y = bandw(kb - m + 1);
            acc = __builtin_amdgcn_wmma_f32_16x16x4_f32(
                false, a, false, b, (short)0, acc, false, false);
        }
    } else {
        // Border: symmetric x-reflection.
#pragma unroll
        for (int s = 0; s < 10; ++s) {
            const int kb = 4 * s + half * 2;
            v2f a;
            a.x = hm[(rowbase + (size_t)reflect_idx(x0 - RAD + kb,     W_IMG)) * NJ + c];
            a.y = hm[(rowbase + (size_t)reflect_idx(x0 - RAD + kb + 1, W_IMG)) * NJ + c];
            v2f b;
            b.x = bandw(kb - m);
            b.y = bandw(kb - m + 1);
            acc = __builtin_amdgcn_wmma_f32_16x16x4_f32(
                false, a, false, b, (short)0, acc, false, false);
        }
    }
    // Stores: one base pointer, constant v*W offsets; guard only partial tiles.
    float* dst = tmp + (size_t)c * ((size_t)H_IMG * W_IMG);
    float* p   = dst + (size_t)(y0 + half * 8) * W_IMG + (x0 + m);
    if (y0 + 16 <= H_IMG) {
#pragma unroll
        for (int v = 0; v < 8; ++v) p[(size_t)v * W_IMG] = acc[v];
    } else {
#pragma unroll
        for (int v = 0; v < 8; ++v)
            if (y0 + half * 8 + v < H_IMG) p[(size_t)v * W_IMG] = acc[v];
    }
}

// ---------------------------------------------------------------------------
// Vertical blur: 4 waves per workgroup, 64x16 output region. Wave 0 issues a
// Tensor-DMA load of the (<=88 rows x 16 cols) input window into LDS, waits on
// TENSORcnt, then all waves run 10 x WMMA f32 K=4 fed from LDS.
// ---------------------------------------------------------------------------
__global__ void blur_v_kernel(const float* __restrict__ tmp, float* __restrict__ hmf) {
    __shared__ float smem[VROWS * 16];
    const int tid  = threadIdx.x;       // 0..127
    const int wave = tid >> 5;
    const int lane = tid & 31;
    const int m    = lane & 15;
    const int half = lane >> 4;
    const int x0  = blockIdx.x * 16;
    const int y0g = blockIdx.y * 64;
    const int c   = blockIdx.z;
    const size_t plane = (size_t)c * ((size_t)H_IMG * W_IMG);

    const int rs    = max(y0g - RAD, 0);
    const int re    = min(y0g + 76, H_IMG);
    const int nrows = re - rs;          // <= 88

#if defined(HAVE_TDM)
    if (wave == 0) {
        const unsigned long long gaddr =
            (unsigned long long)(uintptr_t)(tmp + plane + (size_t)rs * W_IMG + x0);
        const unsigned lds_addr = (unsigned)(uintptr_t)smem;  // low 32b = LDS offset
        v4u g0;
        g0.x = 1u;                                   // count=1, user descriptor
        g0.y = lds_addr;                             // D#.lds_addr
        g0.z = (unsigned)(gaddr & 0xffffffffull);    // global_addr[31:0]
        g0.w = (unsigned)((gaddr >> 32) & 0x1ffffffull) | (2u << 30);  // type=2
        v8i g1;
        g1[0] = (int)(2u << 16);                     // data_size = 4B
        g1[1] = (int)(16u << 16);                    // tensor_dim0 = 16 (low16<<16)
        g1[2] = (int)(((unsigned)nrows & 0xffffu) << 16);  // tensor_dim1 = nrows
        g1[3] = (int)(16u << 16);                    // tile_dim0 = 16
        g1[4] = (int)((unsigned)nrows & 0xffffu);    // tile_dim1 = nrows
        g1[5] = (int)W_IMG;                          // tensor_dim0_stride = W
        g1[6] = 0;
        g1[7] = 0;
        const v4i gz = {0, 0, 0, 0};
        TDM_LOAD(g0, g1, gz, gz);
        __builtin_amdgcn_s_wait_tensorcnt((short)0);
    }
#else
    for (int i = tid; i < nrows * 16; i += 128) {
        const int r = i >> 4, col = i & 15;
        smem[i] = tmp[plane + (size_t)(rs + r) * W_IMG + (x0 + col)];
    }
#endif
    __syncthreads();

    const int y0 = y0g + wave * 16;
    v8f acc = {};
    if (y0g >= RAD && y0g + 76 <= H_IMG) {
        // Interior: reflect is identity; LDS reads use constant offsets s*64(+16).
        const float* base = smem + (wave * 16 + half * 2) * 16 + m;
#pragma unroll
        for (int s = 0; s < 10; ++s) {
            const int kb = 4 * s + half * 2;
            v2f a;
            a.x = bandw(kb - m);
            a.y = bandw(kb - m + 1);
            v2f b;
            b.x = base[s * 64];
            b.y = base[s * 64 + 16];
            acc = __builtin_amdgcn_wmma_f32_16x16x4_f32(
                false, a, false, b, (short)0, acc, false, false);
        }
    } else {
        // Border groups: symmetric y-reflection into the LDS window.
#pragma unroll
        for (int s = 0; s < 10; ++s) {
            const int kb = 4 * s + half * 2;
            v2f a;
            a.x = bandw(kb - m);
            a.y = bandw(kb - m + 1);
            const int r0 = reflect_idx(y0 - RAD + kb,     H_IMG) - rs;
            const int r1 = reflect_idx(y0 - RAD + kb + 1, H_IMG) - rs;
            v2f b;
            b.x = smem[r0 * 16 + m];
            b.y = smem[r1 * 16 + m];
            acc = __builtin_amdgcn_wmma_f32_16x16x4_f32(
                false, a, false, b, (short)0, acc, false, false);
        }
    }
    float* dst = hmf + plane;
    float* p   = dst + (size_t)(y0 + half * 8) * W_IMG + (x0 + m);
    if (y0 + 16 <= H_IMG) {
#pragma unroll
        for (int v = 0; v < 8; ++v) p[(size_t)v * W_IMG] = acc[v];
    } else {
#pragma unroll
        for (int v = 0; v < 8; ++v)
            if (y0 + half * 8 + v < H_IMG) p[(size_t)v * W_IMG] = acc[v];
    }
}

// ---------------------------------------------------------------------------
// Stage 1: per-(channel, 2048-pixel segment) peak mask + top-32 extraction.
// Key = (f32 bits of unblurred value << 32) | pixel index; 0 = empty.
// ---------------------------------------------------------------------------
__global__ void peaks_stage1(const float* __restrict__ hm,
                             const float* __restrict__ hmf,
                             unsigned long long* __restrict__ cand) {
    __shared__ unsigned long long keys[SEG_PIX];
    __shared__ unsigned long long red[256];
    const int t   = threadIdx.x;
    const int seg = blockIdx.x;
    const int c   = blockIdx.y;
    const int HW  = H_IMG * W_IMG;
    const float* plane = hmf + (size_t)c * HW;
    const int p0 = seg * SEG_PIX;

#pragma unroll
    for (int i = 0; i < 8; ++i) {
        const int off = t + i * 256;
        const int p   = p0 + off;
        unsigned long long key = 0ull;
        if (p < HW) {
            const int y = p / W_IMG;
            const int x = p - y * W_IMG;
            const float v = plane[p];
            if (v > 0.1f) {
                const float nu = (y > 0)         ? plane[p - W_IMG] : 0.0f;
                const float nd = (y < H_IMG - 1) ? plane[p + W_IMG] : 0.0f;
                const float nl = (x > 0)         ? plane[p - 1]     : 0.0f;
                const float nr = (x < W_IMG - 1) ? plane[p + 1]     : 0.0f;
                if (v >= nu && v >= nd && v >= nl && v >= nr) {
                    const unsigned int bits = __float_as_uint(hm[(size_t)p * NJ + c]);
                    key = ((unsigned long long)bits << 32) | (unsigned int)p;
                }
            }
        }
        keys[off] = key;
    }
    __syncthreads();

    unsigned long long* out = cand + ((size_t)c * NSEG + seg) * KTOP;
    for (int rnd = 0; rnd < KTOP; ++rnd) {
        unsigned long long mx = 0ull;
#pragma unroll
        for (int i = 0; i < 8; ++i) {
            const unsigned long long k = keys[t + i * 256];
            if (k > mx) mx = k;
        }
        red[t] = mx;
        __syncthreads();
        for (int s = 128; s > 0; s >>= 1) {
            if (t < s && red[t + s] > red[t]) red[t] = red[t + s];
            __syncthreads();
        }
        const unsigned long long win = red[0];
        __syncthreads();
        if (t == 0) out[rnd] = win;
        if (win) {
#pragma unroll
            for (int i = 0; i < 8; ++i) {
                const int off = t + i * 256;
                if (keys[off] == win) keys[off] = 0ull;
            }
        }
        __syncthreads();
    }
}

// ---------------------------------------------------------------------------
// Stage 2: merge 1013*32 candidates per channel -> global top-32.
// Whole candidate set (259 KB) staged in CDNA5's 320 KB LDS.
// ---------------------------------------------------------------------------
__global__ void peaks_stage2(const unsigned long long* __restrict__ cand,
                             float* __restrict__ out) {
    __shared__ unsigned long long buf[NCAND];
    __shared__ unsigned long long red[1024];
    const int t = threadIdx.x;
    const int c = blockIdx.x;
    for (int i = t; i < NCAND; i += 1024) buf[i] = cand[(size_t)c * NCAND + i];
    __syncthreads();

    for (int rnd = 0; rnd < KTOP; ++rnd) {
        unsigned long long mx = 0ull;
        for (int i = t; i < NCAND; i += 1024) {
            const unsigned long long k = buf[i];
            if (k > mx) mx = k;
        }
        red[t] = mx;
        __syncthreads();
        for (int s = 512; s > 0; s >>= 1) {
            if (t < s && red[t + s] > red[t]) red[t] = red[t + s];
            __syncthreads();
        }
        const unsigned long long win = red[0];
        __syncthreads();
        if (t == 0) {
            const int   p     = (int)(win & 0xffffffffull);
            const float val   = __uint_as_float((unsigned int)(win >> 32));
            const bool  valid = (win != 0ull);
            float* o = out + ((size_t)c * KTOP + rnd) * 4;
            o[0] = valid ? (float)(p % W_IMG) : 0.0f;
            o[1] = valid ? (float)(p / W_IMG) : 0.0f;
            o[2] = valid ? val : 0.0f;
            o[3] = valid ? 1.0f : 0.0f;
        }
        if (win) {
            for (int i = t; i < NCAND; i += 1024)
                if (buf[i] == win) buf[i] = 0ull;
        }
        __syncthreads();
    }
}

// ---------------------------------------------------------------------------
// Limb scoring: 24 limbs x 32x32 pairs, 10 PAF samples per pair.
// ---------------------------------------------------------------------------
__global__ void limbs_kernel(const float* __restrict__ paf,
                             const float* __restrict__ peaks,
                             float* __restrict__ limb_out) {
    const int l  = blockIdx.x;
    const int ja = LIMB_A[l], jb = LIMB_B[l];
    const int cx = MAPX[l],  cy = MAPY[l];
    __shared__ float pa[KTOP * 4];
    __shared__ float pb[KTOP * 4];
    if (threadIdx.x < KTOP * 4) {
        pa[threadIdx.x] = peaks[(size_t)ja * KTOP * 4 + threadIdx.x];
        pb[threadIdx.x] = peaks[(size_t)jb * KTOP * 4 + threadIdx.x];
    }
    __syncthreads();

    for (int q = threadIdx.x; q < KTOP * KTOP; q += blockDim.x) {
        const int i = q >> 5, j = q & 31;
        const float xA = pa[i * 4 + 0], yA = pa[i * 4 + 1], sA = pa[i * 4 + 2];
        const float xB = pb[j * 4 + 0], yB = pb[j * 4 + 1], sB = pb[j * 4 + 2];
        const bool  vA = pa[i * 4 + 3] > 0.5f, vB = pb[j * 4 + 3] > 0.5f;

        const float dx = xB - xA, dy = yB - yA;
        const float norm = fmaxf(sqrtf(dx * dx + dy * dy), 0.001f);
        const float ux = dx / norm, uy = dy / norm;

        float sum = 0.0f;
        int   cnt = 0;
#pragma unroll
        for (int k = 0; k < 10; ++k) {
            const float tk = (float)k * (1.0f / 9.0f);
            int sx = (int)rintf(xA + tk * dx);
            int sy = (int)rintf(yA + tk * dy);
            sx = min(max(sx, 0), W_IMG - 1);
            sy = min(max(sy, 0), H_IMG - 1);
            const size_t base = ((size_t)sy * W_IMG + sx) * NPAF;
            const float sm = paf[base + cx] * ux + paf[base + cy] * uy;
            sum += sm;
            cnt += (sm > 0.05f) ? 1 : 0;
        }
        const float prior  = sum * 0.1f + fminf(0.5f * (float)H_IMG / norm - 1.0f, 0.0f);
        const bool  pvalid = (cnt > 8) && (prior > 0.0f) && vA && vB;
        const float total  = prior + sA + sB;

        float* o = limb_out + ((size_t)l * KTOP * KTOP + q) * 3;
        o[0] = pvalid ? prior : 0.0f;
        o[1] = pvalid ? total : 0.0f;
        o[2] = pvalid ? 1.0f  : 0.0f;
    }
}

// ---------------------------------------------------------------------------
extern "C" void kernel_launch(void* const* d_in, const int* in_sizes, int n_in,
                              void* d_out, int out_size, void* d_ws, size_t ws_size,
                              hipStream_t stream) {
    const float* hm  = (const float*)d_in[0];   // (1080,1920,26) f32
    const float* paf = (const float*)d_in[1];   // (1080,1920,52) f32
    float* out = (float*)d_out;                 // 25*32*4 + 24*32*32*3 f32
    char*  ws  = (char*)d_ws;

    const size_t HW = (size_t)H_IMG * W_IMG;
    float* tmp = (float*)ws;                                      // 25*HW f32
    float* hmf = (float*)(ws + NCH * HW * sizeof(float));         // 25*HW f32
    unsigned long long* cand =
        (unsigned long long*)(ws + 2 * NCH * HW * sizeof(float)); // 25*NCAND u64

    blur_h_kernel<<<dim3(W_IMG / 16, (H_IMG + 15) / 16, NCH), 32, 0, stream>>>(hm, tmp);
    blur_v_kernel<<<dim3(W_IMG / 16, (H_IMG + 63) / 64, NCH), 128, 0, stream>>>(tmp, hmf);
    peaks_stage1<<<dim3(NSEG, NCH), 256, 0, stream>>>(hm, hmf, cand);
    peaks_stage2<<<NCH, 1024, 0, stream>>>(cand, out);
    limbs_kernel<<<24, 256, 0, stream>>>(paf, out, out + NCH * KTOP * 4);
}